// OctreeAvgPool_72808285602332
// MI455X (gfx1250) — compile-verified
//
#include <hip/hip_runtime.h>
#include <stdint.h>

// OctreeAvgPool: out[r,:] = (1/(cnt_r+1e-8)) * sum_k mask(r,k) * data[neigh[r,k],:]
// data: [N1,128] fp32   neigh: [N2,8] int32   out: [N2,128] fp32
//
// Bandwidth-bound (~1.15 GB @ 23.3 TB/s => ~50us floor). gfx1250 path: the
// Tensor Data Mover's GATHER MODE (one TENSOR_LOAD_TO_LDS fetches the 8
// indexed 512B feature rows = 4KB into LDS), double-buffered per wave via
// TENSORcnt so the DMA of row i+1 overlaps the masked LDS reduction of row i.
// TDM issued via inline asm (portable across ROCm7.2/clang-22 and
// amdgpu-toolchain/clang-23, whose builtin arities differ).

typedef float        f32x4 __attribute__((ext_vector_type(4)));
typedef unsigned int u32x4 __attribute__((ext_vector_type(4)));
typedef int          i32x4 __attribute__((ext_vector_type(4)));
typedef int          i32x8 __attribute__((ext_vector_type(8)));

#define CCH 128            // channels per row (512 bytes)
#define KN  8              // neighbors per output row
#define WPB 8              // waves per 256-thread block
#define TILEROWS 8         // rows per TDM gather tile
#define BUFFLOATS (TILEROWS * CCH)   // 4KB per buffer

#if defined(__gfx1250__)
#define USE_TDM 1
#else
#define USE_TDM 0          // host parse / other targets: direct-gather body
#endif

__global__ __launch_bounds__(256) void OctreeAvgPool_gfx1250_kernel(
    const float* __restrict__ data,
    const int*   __restrict__ neigh,
    float*       __restrict__ out,
    int n2, int n1)
{
  const int lane = (int)(threadIdx.x & 31u);
  const int wave = __builtin_amdgcn_readfirstlane((int)(threadIdx.x >> 5));
  const int wpb  = (int)(blockDim.x >> 5);
  const int stride = (int)gridDim.x * wpb;
  int row = (int)blockIdx.x * wpb + wave;

#if USE_TDM
  __shared__ float buf[WPB][2][BUFFLOATS];   // 2 x 4KB per wave = 64KB/block
  if (row >= n2) return;

  // ---- Loop-invariant descriptor pieces (CDNA5 ISA) ----
  const uint64_t ga   = (uint64_t)(uintptr_t)data;
  const uint32_t gaLo = (uint32_t)ga;
  const uint32_t gaHi = (uint32_t)((ga >> 32) & 0x01FFFFFFull) | (2u << 30); // type=2 ("image")
  // group0: count=1 | gather_index_size=32b (bit30) | gather_mode (bit31);
  //         lds_addr (low 32 bits of flat shared ptr == group-LDS byte offset);
  //         57-bit global_addr = tensor base (row indices select rows).
  const u32x4 g0a = { 0xC0000001u, (uint32_t)(uintptr_t)&buf[wave][0][0], gaLo, gaHi };
  const u32x4 g0b = { 0xC0000001u, (uint32_t)(uintptr_t)&buf[wave][1][0], gaLo, gaHi };
  // group1: wg_mask=0 (not in cluster), data_size=2 (4B) at bits[17:16];
  //         tensor_dim0=128, tensor_dim1=N1 (OOB bound), tile_dim0=128,
  //         tile_dim1=#indices=8, tensor_dim0_stride=128. dim1_stride ignored
  //         in gather mode.
  const i32x8 g1 = {
      (int)0x00020000u,                                        // data_size=4B
      (int)(128u << 16),                                       // tensor_dim0 lo16
      (int)(((uint32_t)n1 & 0xFFFFu) << 16),                   // dim0 hi16=0 | tensor_dim1 lo16
      (int)((((uint32_t)n1 >> 16) & 0xFFFFu) | (128u << 16)),  // tensor_dim1 hi16 | tile_dim0=128
      TILEROWS,                                                // tile_dim1 = 8 indices
      CCH,                                                     // tensor_dim0_stride lo32 = 128
      0, 0 };

#define TDM_GATHER(G0, GI2, GI3)                                         \
  asm volatile("tensor_load_to_lds %0, %1, %2, %3"                       \
               :                                                         \
               : "s"(G0), "s"(g1), "s"(GI2), "s"(GI3)                    \
               : "memory")

  int idc[KN];
  int idn[KN] = {0,0,0,0,0,0,0,0};

  // Prologue: load row's indices (uniform -> SGPRs) and kick TDM into buf0.
  {
    const int* nbp = neigh + (size_t)row * KN;
#pragma unroll
    for (int k = 0; k < KN; ++k) idc[k] = __builtin_amdgcn_readfirstlane(nbp[k]);
    const i32x4 gi2 = { idc[0] < 0 ? 0 : idc[0], idc[1] < 0 ? 0 : idc[1],
                        idc[2] < 0 ? 0 : idc[2], idc[3] < 0 ? 0 : idc[3] };
    const i32x4 gi3 = { idc[4] < 0 ? 0 : idc[4], idc[5] < 0 ? 0 : idc[5],
                        idc[6] < 0 ? 0 : idc[6], idc[7] < 0 ? 0 : idc[7] };
    TDM_GATHER(g0a, gi2, gi3);
  }

  int cur = 0;
  for (; row < n2; row += stride) {
    const int nxt = row + stride;
    if (nxt < n2) {
      // Software pipeline: DMA next row's gather into the other buffer, then
      // wait only for the OLDEST tensor op (in-order per wave) => overlap.
      __builtin_prefetch(neigh + (size_t)(nxt + stride) * KN, 0, 1);
      const int* nbp = neigh + (size_t)nxt * KN;
#pragma unroll
      for (int k = 0; k < KN; ++k) idn[k] = __builtin_amdgcn_readfirstlane(nbp[k]);
      const i32x4 gi2 = { idn[0] < 0 ? 0 : idn[0], idn[1] < 0 ? 0 : idn[1],
                          idn[2] < 0 ? 0 : idn[2], idn[3] < 0 ? 0 : idn[3] };
      const i32x4 gi3 = { idn[4] < 0 ? 0 : idn[4], idn[5] < 0 ? 0 : idn[5],
                          idn[6] < 0 ? 0 : idn[6], idn[7] < 0 ? 0 : idn[7] };
      asm volatile("s_wait_dscnt 0x0" ::: "memory");   // LDS reads of 2-ago iter done
      if (cur) { TDM_GATHER(g0a, gi2, gi3); } else { TDM_GATHER(g0b, gi2, gi3); }
      asm volatile("s_wait_tensorcnt 0x1" ::: "memory");
    } else {
      asm volatile("s_wait_tensorcnt 0x0" ::: "memory");
    }

    // Masked mean of the 8 gathered rows now sitting in LDS.
    float cnt = 0.0f;
#pragma unroll
    for (int k = 0; k < KN; ++k) cnt += (idc[k] >= 0) ? 1.0f : 0.0f;
    const float inv = 1.0f / (cnt + 1e-8f);

    const float* wb = &buf[wave][cur][0];
    f32x4 acc = {0.0f, 0.0f, 0.0f, 0.0f};
#pragma unroll
    for (int r = 0; r < TILEROWS; ++r) {
      const float m = (idc[r] >= 0) ? 1.0f : 0.0f;     // scalar mask
      const f32x4 v = *(const f32x4*)(wb + r * CCH + (lane << 2));  // ds_load_b128
      const f32x4 mv = {m, m, m, m};
      acc += v * mv;
    }
    const f32x4 sv = {inv, inv, inv, inv};
    acc *= sv;
    __builtin_nontemporal_store(acc, (f32x4*)(out + (size_t)row * CCH) + lane);

#pragma unroll
    for (int k = 0; k < KN; ++k) idc[k] = idn[k];
    cur ^= 1;
  }
#undef TDM_GATHER
#else
  // -------- Fallback: direct VGPR gather (round-1 kernel) ----------
  (void)n1;
  for (; row < n2; row += stride) {
    const int* nb = neigh + (size_t)row * KN;
    __builtin_prefetch(neigh + (size_t)(row + stride) * KN, 0, 1);

    int idx[KN];
#pragma unroll
    for (int k = 0; k < KN; ++k) idx[k] = __builtin_amdgcn_readfirstlane(nb[k]);

    float cnt = 0.0f;
#pragma unroll
    for (int k = 0; k < KN; ++k) cnt += (idx[k] >= 0) ? 1.0f : 0.0f;
    const float inv = 1.0f / (cnt + 1e-8f);

    f32x4 acc = {0.0f, 0.0f, 0.0f, 0.0f};
#pragma unroll
    for (int k = 0; k < KN; ++k) {
      const int   safe = (idx[k] < 0) ? 0 : idx[k];
      const float m    = (idx[k] < 0) ? 0.0f : 1.0f;
      const float* base = data + (size_t)safe * CCH;
      const f32x4 v = *(const f32x4*)(base + (lane << 2));
      const f32x4 mv = {m, m, m, m};
      acc += v * mv;
    }
    const f32x4 sv = {inv, inv, inv, inv};
    acc *= sv;
    __builtin_nontemporal_store(acc, (f32x4*)(out + (size_t)row * CCH) + lane);
  }
#endif
}

extern "C" void kernel_launch(void* const* d_in, const int* in_sizes, int n_in,
                              void* d_out, int out_size, void* d_ws, size_t ws_size,
                              hipStream_t stream) {
  const float* data  = (const float*)d_in[0];   // [N1, 128] fp32
  const int*   neigh = (const int*)d_in[1];     // [N2, 8] int32
  float*       out   = (float*)d_out;           // [N2, 128] fp32

  const int n2 = in_sizes[1] / KN;
  const int n1 = in_sizes[0] / CCH;

  // 256 threads = 8 wave32s/block; ~8 rows per wave so the TDM double-buffer
  // pipeline has depth while keeping tens of thousands of waves in flight.
  const int rowsPerWave = 8;
  int blocks = (n2 + WPB * rowsPerWave - 1) / (WPB * rowsPerWave);
  if (blocks < 1) blocks = 1;

  OctreeAvgPool_gfx1250_kernel<<<blocks, 256, 0, stream>>>(data, neigh, out, n2, n1);
}